// MemoryModule_40192303956223
// MI455X (gfx1250) — compile-verified
//
#include <hip/hip_runtime.h>
#include <math.h>
#include <stdint.h>

typedef __attribute__((ext_vector_type(2))) float v2f;
typedef __attribute__((ext_vector_type(4))) float v4f;
typedef __attribute__((ext_vector_type(8))) float v8f;

#define BT 16384   // B*T
#define DD 1024    // D
#define MMEM 4096  // M memory slots
#define KTOP 8

// ---------------------------------------------------------------------------
// CDNA5 async global->LDS copy (16B per lane), tracked by ASYNCcnt.
// Generic pointer to __shared__ truncates to the LDS byte offset (LDS
// aperture: addr[31:0] = offset), which is exactly what VDST must hold.
// ---------------------------------------------------------------------------
__device__ __forceinline__ void async_g2l_b128(float* lds_dst, const float* gsrc) {
  uint32_t l = (uint32_t)(uintptr_t)lds_dst;
  asm volatile("global_load_async_to_lds_b128 %0, %1, off"
               :: "v"(l), "v"(gsrc)
               : "memory");
}
__device__ __forceinline__ void wait_async0() {
  asm volatile("s_wait_asynccnt 0x0" ::: "memory");
}

// ---------------------------------------------------------------------------
// fp32 WMMA GEMM: C[M,N] = A[M,K] @ B[K,N], row-major.
// Block = 256 threads = 8 waves covering a 64x128 output tile
// (wave layout 4 rows x 2 cols, each wave 16x64 via 4 accumulators).
// A-frag amortized over 4 WMMAs; block-level re-read traffic ~1.5 GB/GEMM.
// ---------------------------------------------------------------------------
__global__ __launch_bounds__(256) void gemm_f32(const float* __restrict__ A,
                                                const float* __restrict__ Bm,
                                                float* __restrict__ C,
                                                int N, int K) {
  const int lane = threadIdx.x & 31;
  const int wave = threadIdx.x >> 5;
  const int rowbase = blockIdx.y * 64 + (wave >> 1) * 16;
  const int colbase = blockIdx.x * 128 + (wave & 1) * 64;
  const int r = lane & 15;
  const int koff = (lane >> 4) << 1;

  const float* ap = A + (size_t)(rowbase + r) * K + koff;
  const float* bp = Bm + (size_t)koff * N + colbase + r;

  v8f acc[4] = {};
  for (int k = 0; k < K; k += 4) {
    v2f a = *(const v2f*)(ap + k);
    size_t off = (size_t)k * N;
#pragma unroll
    for (int c = 0; c < 4; ++c) {
      v2f b;
      b.x = bp[off + c * 16];
      b.y = bp[off + N + c * 16];
      acc[c] = __builtin_amdgcn_wmma_f32_16x16x4_f32(false, a, false, b, (short)0, acc[c], false, false);
    }
  }

  const int hi8 = (lane >> 4) << 3;
#pragma unroll
  for (int c = 0; c < 4; ++c) {
    int col = colbase + c * 16 + r;
#pragma unroll
    for (int v = 0; v < 8; ++v)
      C[(size_t)(rowbase + hi8 + v) * N + col] = acc[c][v];
  }
}

// ---------------------------------------------------------------------------
// sim + top-8 fused. WMMA M-dim = memory slots (A = keys), N-dim = tokens
// (B = Q^T). Block = 4 waves / 64 tokens. Whole Q tile (256 KB) async-staged
// into LDS ONCE; 16-row keys tile (64 KB) async-staged per m-tile.
// Total LDS = 320 KB (CDNA5 per-workgroup max). Q L2 traffic: 64 MB total.
// Lane l and l^16 split the 4096 slots for token (l&15); private sorted
// top-8 each, merged with one shfl pass, then softmax.
// ---------------------------------------------------------------------------
__global__ __launch_bounds__(128) void simtopk(const float* __restrict__ Q,
                                               const float* __restrict__ keys,
                                               float* __restrict__ topw,
                                               int* __restrict__ topi) {
  __shared__ __align__(16) float sQ[64 * DD];     // 256 KB
  __shared__ __align__(16) float sKeys[16 * DD];  // 64 KB
  const int tid = threadIdx.x;
  const int lane = tid & 31;
  const int wave = tid >> 5;
  const int tokbase = blockIdx.x * 64 + wave * 16;
  const int r = lane & 15;
  const int koff = (lane >> 4) << 1;

  // stage the block's Q tile (64 tokens, contiguous rows) via async DMA
  {
    const float* gq = Q + (size_t)blockIdx.x * 64 * DD + tid * 512;
    float* dq = sQ + tid * 512;
#pragma unroll 4
    for (int i = 0; i < 128; ++i)
      async_g2l_b128(dq + i * 4, gq + i * 4);
  }

  const float* sA = sKeys + r * DD + koff;                 // keys frag base
  const float* sB = sQ + (wave * 16 + r) * DD + koff;      // Q^T frag base

  float tv[8];
  int ti[8];
#pragma unroll
  for (int j = 0; j < 8; ++j) { tv[j] = -3.4e38f; ti[j] = 0; }

  for (int mt = 0; mt < MMEM / 16; ++mt) {
    __syncthreads();  // everyone done reading previous sKeys
    {
      const float* gk = keys + (size_t)mt * (16 * DD) + tid * 128;
      float* dk = sKeys + tid * 128;
#pragma unroll 8
      for (int i = 0; i < 32; ++i)
        async_g2l_b128(dk + i * 4, gk + i * 4);
    }
    if (mt + 1 < MMEM / 16)
      __builtin_prefetch(keys + (size_t)(mt + 1) * (16 * DD) + tid * 128, 0, 0);
    wait_async0();    // own wave's async copies (incl. sQ on first iter) done
    __syncthreads();  // all waves' copies visible

    v8f acc = {};
#pragma unroll 4
    for (int k = 0; k < DD; k += 4) {
      v2f a = *(const v2f*)(sA + k);
      v2f b = *(const v2f*)(sB + k);
      acc = __builtin_amdgcn_wmma_f32_16x16x4_f32(false, a, false, b, (short)0, acc, false, false);
    }

    const int ibase = mt * 16 + ((lane >> 4) << 3);
#pragma unroll
    for (int v = 0; v < 8; ++v) {
      float c = acc[v];
      if (c > tv[7]) {
        tv[7] = c; ti[7] = ibase + v;
#pragma unroll
        for (int j = 7; j > 0; --j) {
          if (tv[j] > tv[j - 1]) {
            float tf = tv[j]; tv[j] = tv[j - 1]; tv[j - 1] = tf;
            int tx = ti[j]; ti[j] = ti[j - 1]; ti[j - 1] = tx;
          }
        }
      }
    }
  }

  // merge the two half-lists (lane pair l, l^16) -> exact top-8 per token
  float ov[8];
  int oi[8];
#pragma unroll
  for (int j = 0; j < 8; ++j) {
    ov[j] = __shfl(tv[j], lane ^ 16, 32);
    oi[j] = __shfl(ti[j], lane ^ 16, 32);
  }
  float mv[8];
  int mi[8];
  int p = 0, q = 0;
#pragma unroll
  for (int j = 0; j < 8; ++j) {
    bool takeT = (q >= 8) || ((p < 8) && (tv[p] >= ov[q]));
    if (takeT) { mv[j] = tv[p]; mi[j] = ti[p]; ++p; }
    else       { mv[j] = ov[q]; mi[j] = oi[q]; ++q; }
  }

  // softmax over 8 with 1/sqrt(D) = 1/32 folded in
  float vmax = mv[0];
  float s = 0.f, w[8];
#pragma unroll
  for (int j = 0; j < 8; ++j) { w[j] = expf((mv[j] - vmax) * 0.03125f); s += w[j]; }
  float inv = 1.f / s;

  if (lane < 16) {
    int t = tokbase + r;
#pragma unroll
    for (int j = 0; j < 8; ++j) {
      topw[t * KTOP + j] = w[j] * inv;
      topi[t * KTOP + j] = mi[j];
    }
  }
}

// ---------------------------------------------------------------------------
// retrieved_raw[t,:] = sum_k w[t,k] * V[idx[t,k],:]  (one block per token;
// V is 16 MB -> L2 resident)
// ---------------------------------------------------------------------------
__global__ __launch_bounds__(256) void gather_wsum(const float* __restrict__ topw,
                                                   const int* __restrict__ topi,
                                                   const float* __restrict__ V,
                                                   float* __restrict__ out) {
  const int t = blockIdx.x;
  const int d0 = threadIdx.x * 4;
  v4f acc = {};
#pragma unroll
  for (int k = 0; k < KTOP; ++k) {
    float wk = topw[t * KTOP + k];
    int id = topi[t * KTOP + k];
    v4f vv = *(const v4f*)(V + (size_t)id * DD + d0);
    acc += wk * vv;
  }
  *(v4f*)(out + (size_t)t * DD + d0) = acc;
}

// ---------------------------------------------------------------------------
// h = gelu([x | retrieved] @ gW1 + gb1); K=2048 split as two 1024 loops over
// two A pointers (no concat buffer). Same 64x128 block tiling as gemm_f32,
// fused bias + exact-erf GELU.
// ---------------------------------------------------------------------------
__global__ __launch_bounds__(256) void gemm_concat_gelu(const float* __restrict__ A1,
                                                        const float* __restrict__ A2,
                                                        const float* __restrict__ Bm,
                                                        const float* __restrict__ bias,
                                                        float* __restrict__ C,
                                                        int N) {
  const int lane = threadIdx.x & 31;
  const int wave = threadIdx.x >> 5;
  const int rowbase = blockIdx.y * 64 + (wave >> 1) * 16;
  const int colbase = blockIdx.x * 128 + (wave & 1) * 64;
  const int r = lane & 15;
  const int koff = (lane >> 4) << 1;

  const float* ap1 = A1 + (size_t)(rowbase + r) * DD + koff;
  const float* ap2 = A2 + (size_t)(rowbase + r) * DD + koff;
  const float* bp  = Bm + (size_t)koff * N + colbase + r;

  v8f acc[4] = {};
  for (int k = 0; k < DD; k += 4) {      // K 0..1023: A = x
    v2f a = *(const v2f*)(ap1 + k);
    size_t off = (size_t)k * N;
#pragma unroll
    for (int c = 0; c < 4; ++c) {
      v2f b;
      b.x = bp[off + c * 16];
      b.y = bp[off + N + c * 16];
      acc[c] = __builtin_amdgcn_wmma_f32_16x16x4_f32(false, a, false, b, (short)0, acc[c], false, false);
    }
  }
  for (int k = 0; k < DD; k += 4) {      // K 1024..2047: A = retrieved
    v2f a = *(const v2f*)(ap2 + k);
    size_t off = (size_t)(DD + k) * N;
#pragma unroll
    for (int c = 0; c < 4; ++c) {
      v2f b;
      b.x = bp[off + c * 16];
      b.y = bp[off + N + c * 16];
      acc[c] = __builtin_amdgcn_wmma_f32_16x16x4_f32(false, a, false, b, (short)0, acc[c], false, false);
    }
  }

  const int hi8 = (lane >> 4) << 3;
#pragma unroll
  for (int c = 0; c < 4; ++c) {
    int col = colbase + c * 16 + r;
#pragma unroll
    for (int v = 0; v < 8; ++v) {
      float val = acc[c][v] + bias[col];
      val = 0.5f * val * (1.f + erff(val * 0.70710678118f));
      C[(size_t)(rowbase + hi8 + v) * N + col] = val;
    }
  }
}

// ---------------------------------------------------------------------------
// gate = sigmoid(h[t] . gW2 + gb2); out = x + gate * retrieved
// ---------------------------------------------------------------------------
__global__ __launch_bounds__(256) void gate_out(const float* __restrict__ h,
                                                const float* __restrict__ gW2,
                                                const float* __restrict__ gb2,
                                                const float* __restrict__ x,
                                                const float* __restrict__ retr,
                                                float* __restrict__ out) {
  __shared__ float red[256];
  __shared__ float gateS;
  const int t = blockIdx.x;
  const int tid = threadIdx.x;
  float p = h[(size_t)t * 512 + tid] * gW2[tid] +
            h[(size_t)t * 512 + tid + 256] * gW2[tid + 256];
  red[tid] = p;
  __syncthreads();
  for (int s = 128; s > 0; s >>= 1) {
    if (tid < s) red[tid] += red[tid + s];
    __syncthreads();
  }
  if (tid == 0) gateS = 1.f / (1.f + expf(-(red[0] + gb2[0])));
  __syncthreads();
  float g = gateS;
  const int d0 = tid * 4;
  v4f xv = *(const v4f*)(x + (size_t)t * DD + d0);
  v4f rv = *(const v4f*)(retr + (size_t)t * DD + d0);
  *(v4f*)(out + (size_t)t * DD + d0) = xv + g * rv;
}

// ---------------------------------------------------------------------------
extern "C" void kernel_launch(void* const* d_in, const int* in_sizes, int n_in,
                              void* d_out, int out_size, void* d_ws, size_t ws_size,
                              hipStream_t stream) {
  const float* x    = (const float*)d_in[0];
  const float* keys = (const float*)d_in[1];
  const float* vals = (const float*)d_in[2];
  const float* Wq   = (const float*)d_in[3];
  const float* Wo   = (const float*)d_in[4];
  const float* gW1  = (const float*)d_in[5];
  const float* gb1  = (const float*)d_in[6];
  const float* gW2  = (const float*)d_in[7];
  const float* gb2  = (const float*)d_in[8];
  float* out = (float*)d_out;

  // workspace layout (floats): ~161 MB
  float* Qbuf  = (float*)d_ws;                     // [16384,1024] Q, reused as retrieved_raw
  float* retrF = Qbuf  + (size_t)BT * DD;          // [16384,1024] retrieved @ Wo
  float* hbuf  = retrF + (size_t)BT * DD;          // [16384,512]
  float* topw  = hbuf  + (size_t)BT * 512;         // [16384,8]
  int*   topi  = (int*)(topw + (size_t)BT * KTOP); // [16384,8]

  // 1. Q = x @ Wq
  gemm_f32<<<dim3(DD / 128, BT / 64), 256, 0, stream>>>(x, Wq, Qbuf, DD, DD);
  // 2. sim + top-8 + softmax weights (fp32 WMMA, async-LDS staged)
  simtopk<<<dim3(BT / 64), 128, 0, stream>>>(Qbuf, keys, topw, topi);
  // 3. weighted gather of values (overwrites Qbuf; Q is dead)
  gather_wsum<<<dim3(BT), 256, 0, stream>>>(topw, topi, vals, Qbuf);
  // 4. retrieved = retrieved_raw @ Wo
  gemm_f32<<<dim3(DD / 128, BT / 64), 256, 0, stream>>>(Qbuf, Wo, retrF, DD, DD);
  // 5. h = gelu([x | retrieved] @ gW1 + gb1)
  gemm_concat_gelu<<<dim3(512 / 128, BT / 64), 256, 0, stream>>>(x, retrF, gW1, gb1, hbuf, 512);
  // 6. gate + residual output
  gate_out<<<dim3(BT), 256, 0, stream>>>(hbuf, gW2, gb2, x, retrF, out);
}